// LLT_82978768159405
// MI455X (gfx1250) — compile-verified
//
#include <hip/hip_runtime.h>

typedef __attribute__((ext_vector_type(2))) float v2f;
typedef __attribute__((ext_vector_type(8))) float v8f;

// One V_WMMA_F32_16X16X4_F32 per 4 edges:
//   A (16x4)  = 4 stacked 4x4 edge blocks (optionally transposed at load)
//   B (4x16)  = gathered 4-vectors in columns 0,4,8,12 (harmless dups elsewhere)
//   D (16x16) : result for edge eL at D[4*eL + r][4*eL]
//
// A-operand f32 16x4 lane layout (ISA 7.12.2):
//   lanes 0-15 : M = lane,    VGPR0=K0, VGPR1=K1
//   lanes16-31 : M = lane-16, VGPR0=K2, VGPR1=K3
// B-operand 4x16 is transpose-symmetric (lane n holds column n, K split by half).
// D layout: VGPR v -> M=v (lanes 0-15) / M=v+8 (lanes 16-31), N = lane%16.
// => results live in c[0..3]@lane0 (e0), c[4..7]@lane4 (e1),
//                    c[0..3]@lane24 (e2), c[4..7]@lane28 (e3).
template <bool TRANSPOSE>
__global__ __launch_bounds__(256) void llt_phase_kernel(
    const float* __restrict__ x_in,       // [N,4] gathered vectors
    const int* __restrict__ gather_idx,   // [E]
    const int* __restrict__ scatter_idx,  // [E]
    const float* __restrict__ blocks,     // [E,4,4] row-major
    float* __restrict__ y_out,            // [N,4] atomically accumulated
    int n_edges, int edges_per_wave)
{
    const unsigned lane = threadIdx.x & 31u;
    const unsigned wave = (blockIdx.x * blockDim.x + threadIdx.x) >> 5;

    const unsigned m    = lane & 15u;  // 0..15 : stacked-matrix row
    const unsigned half = lane >> 4;   // 0/1   : K half
    const unsigned kb   = half * 2u;   // K base: 0 or 2
    const unsigned eL   = m >> 2;      // local edge 0..3
    const unsigned i    = m & 3u;      // row inside the 4x4 block

    const bool act =
        (lane == 0u) | (lane == 4u) | (lane == 24u) | (lane == 28u);
    const bool hi = (lane & 4u) != 0u;

    int et = (int)(wave * (unsigned)edges_per_wave);  // first edge, uniform
    if (et >= n_edges) return;
    int et_end = et + edges_per_wave;
    if (et_end > n_edges) et_end = n_edges;

    for (; et < et_end; et += 4) {  // 4 edges per WMMA, loop is wave-uniform
        int e = et + (int)eL;
        const bool evalid = (e < n_edges);
        if (!evalid) e = n_edges - 1;  // clamp: OOB lanes read safely

        // ---- A operand: 4 stacked blocks; wave covers 256 contiguous B ----
        const float* blk = blocks + (unsigned)e * 16u;
        v2f a;
        if (TRANSPOSE) {
            a.x = blk[(kb + 0u) * 4u + i];  // A^T[i][k] = A[k][i]
            a.y = blk[(kb + 1u) * 4u + i];
        } else {
            a = *(const v2f*)(blk + i * 4u + kb);  // contiguous float2
        }

        // ---- B operand: gathered vector for this lane's edge ----
        const unsigned v = (unsigned)gather_idx[e];
        const v2f b = *(const v2f*)(x_in + v * 4u + kb);

        v8f c = {};
        // 8 args: (neg_a, A, neg_b, B, c_mod, C, reuse_a, reuse_b)
        c = __builtin_amdgcn_wmma_f32_16x16x4_f32(false, a, false, b,
                                                  (short)0, c, false, false);

        // ---- extract: constant-index extracts, then opaque barrier so
        //      InstCombine cannot refold the selects into dynamic extracts ----
        float lo0 = c[0], lo1 = c[1], lo2 = c[2], lo3 = c[3];
        float h0 = c[4], h1 = c[5], h2 = c[6], h3 = c[7];
        asm("" : "+v"(lo0), "+v"(lo1), "+v"(lo2), "+v"(lo3),
                 "+v"(h0), "+v"(h1), "+v"(h2), "+v"(h3));
        const float r0 = hi ? h0 : lo0;   // plain v_cndmask each
        const float r1 = hi ? h1 : lo1;
        const float r2 = hi ? h2 : lo2;
        const float r3 = hi ? h3 : lo3;

        if (act && evalid) {
            const unsigned d = (unsigned)scatter_idx[e];
            float* dst = y_out + d * 4u;
            __hip_atomic_fetch_add(dst + 0, r0, __ATOMIC_RELAXED,
                                   __HIP_MEMORY_SCOPE_AGENT);
            __hip_atomic_fetch_add(dst + 1, r1, __ATOMIC_RELAXED,
                                   __HIP_MEMORY_SCOPE_AGENT);
            __hip_atomic_fetch_add(dst + 2, r2, __ATOMIC_RELAXED,
                                   __HIP_MEMORY_SCOPE_AGENT);
            __hip_atomic_fetch_add(dst + 3, r3, __ATOMIC_RELAXED,
                                   __HIP_MEMORY_SCOPE_AGENT);
        }
    }
}

extern "C" void kernel_launch(void* const* d_in, const int* in_sizes, int n_in,
                              void* d_out, int out_size, void* d_ws,
                              size_t ws_size, hipStream_t stream)
{
    const float* x          = (const float*)d_in[0];  // [N,4] f32
    const int*   edge_index = (const int*)d_in[1];    // [2,E] int32
    const float* boo        = (const float*)d_in[2];  // [E,4,4] f32

    const int n_nodes = in_sizes[0] / 4;
    const int n_edges = in_sizes[1] / 2;

    const int* row = edge_index;            // edge_index[0]
    const int* col = edge_index + n_edges;  // edge_index[1]

    float* out  = (float*)d_out;
    float* lt_x = (float*)d_ws;  // [N,4] scratch = L^T x

    hipMemsetAsync(lt_x, 0, (size_t)n_nodes * 4 * sizeof(float), stream);
    hipMemsetAsync(out, 0, (size_t)n_nodes * 4 * sizeof(float), stream);

    const int edges_per_wave = 16;  // 4 WMMA tiles per wave
    const long long waves =
        ((long long)n_edges + edges_per_wave - 1) / edges_per_wave;
    const int threads       = 256;  // 8 waves / block
    const int waves_per_blk = threads / 32;
    const int blocks_ = (int)((waves + waves_per_blk - 1) / waves_per_blk);

    // Phase 1: lt_x[col] += A^T x[row]   (spmv_t)
    llt_phase_kernel<true><<<blocks_, threads, 0, stream>>>(
        x, row, col, boo, lt_x, n_edges, edges_per_wave);
    // Phase 2: out[row] += A * lt_x[col] (spmv)
    llt_phase_kernel<false><<<blocks_, threads, 0, stream>>>(
        lt_x, col, row, boo, out, n_edges, edges_per_wave);
}